// PositionalEmbedding_6734508720782
// MI455X (gfx1250) — compile-verified
//
#include <hip/hip_runtime.h>

// Tree positional embedding, MI455X (gfx1250, wave32).
//
// out[i, 8j+s] = 1.0 iff j < depth[i] and child_idx[ancestor_j(i)] == s,
// where ancestor_0(i) = i, ancestor_{j+1} = parent[ancestor_j].
//
// Pure store-bandwidth problem: 102.4 MB of output, ~4.5 us floor at
// 23.3 TB/s. One wave32 per 128-float row; the ancestor walk is wave-uniform
// (SMEM scalar loads), the row is staged in LDS and drained with the CDNA5
// async LDS->global store engine (ASYNCcnt path), 512 B per wave.

#define BRANCH 8
#define KDEPTH 16
#define HEMB   (BRANCH * KDEPTH)   // 128 floats = 512 B per row
#define WAVES_PER_BLOCK 4
#define BLOCK_THREADS   (WAVES_PER_BLOCK * 32)

__global__ __launch_bounds__(BLOCK_THREADS)
void tree_pos_emb_kernel(const int* __restrict__ parent,
                         const int* __restrict__ child_idx,
                         const int* __restrict__ depth,
                         float* __restrict__ out,
                         int N)
{
    __shared__ float lds_rows[WAVES_PER_BLOCK * HEMB];  // 2 KB of 320 KB/WGP

    const int lane = threadIdx.x & 31;
    const int wave = threadIdx.x >> 5;

    // Force the row id wave-uniform so the ancestor walk lowers to scalar
    // (SMEM) loads through the constant cache instead of broadcast VMEM.
    const int row = __builtin_amdgcn_readfirstlane(
        (int)(blockIdx.x * WAVES_PER_BLOCK) + wave);
    if (row >= N) return;

    // ---- wave-uniform ancestor walk: pack child nibbles ------------------
    // 0xF nibbles mark "no level here" and can never match a slot in [0,8).
    unsigned long long packed = ~0ull;
    int d = depth[row];
    if (d > KDEPTH) d = KDEPTH;
    int cur = row;
    for (int j = 0; j < d; ++j) {
        unsigned long long c = (unsigned)child_idx[cur] & 0xFu;
        packed = (packed & ~(0xFull << (4 * j))) | (c << (4 * j));
        cur = parent[cur];
    }

    // ---- per-lane: 4 consecutive floats of the 128-float row -------------
    const int      blk  = lane >> 1;         // one-hot block 0..15
    const unsigned s0   = (lane & 1) * 4u;   // first slot this lane covers
    const unsigned nib  = (unsigned)((packed >> (4 * blk)) & 0xFull);
    float4 v;
    v.x = (nib == s0 + 0u) ? 1.0f : 0.0f;
    v.y = (nib == s0 + 1u) ? 1.0f : 0.0f;
    v.z = (nib == s0 + 2u) ? 1.0f : 0.0f;
    v.w = (nib == s0 + 3u) ? 1.0f : 0.0f;

    // Stage the row in LDS: ds_store_b128, 16 B per lane, conflict-free.
    float4* lrow = (float4*)&lds_rows[wave * HEMB];
    lrow[lane] = v;

    // LDS writes must land before the async DMA engine reads them.
    asm volatile("s_wait_dscnt 0" ::: "memory");

    // CDNA5 async LDS->global store: each lane moves one b128 packet,
    // so one instruction streams the whole 512 B row (perfectly coalesced).
    const float4*  gdst = (const float4*)(out + (size_t)row * HEMB) + lane;
    const unsigned lofs = (unsigned)(size_t)&lrow[lane];  // addr[31:0] = LDS offset
    asm volatile("global_store_async_from_lds_b128 %0, %1, off"
                 :: "v"(gdst), "v"(lofs)
                 : "memory");

    // Drain ASYNCcnt before the wave retires (ENDPGM's implicit wait-idle
    // would also cover this; explicit wait documents the dependency).
    asm volatile("s_wait_asynccnt 0" ::: "memory");
}

extern "C" void kernel_launch(void* const* d_in, const int* in_sizes, int n_in,
                              void* d_out, int out_size, void* d_ws, size_t ws_size,
                              hipStream_t stream) {
    // setup_inputs() order: parent, child_idx, depth, p_emb, n, k
    const int* parent = (const int*)d_in[0];
    const int* child  = (const int*)d_in[1];
    const int* depth  = (const int*)d_in[2];
    // d_in[3] (p_emb) unused by the forward; d_in[4]/d_in[5] are the compile-
    // time constants n=8, k=16 baked into HEMB.
    float* out = (float*)d_out;

    const int N = in_sizes[0];
    const int blocks = (N + WAVES_PER_BLOCK - 1) / WAVES_PER_BLOCK;
    tree_pos_emb_kernel<<<blocks, BLOCK_THREADS, 0, stream>>>(
        parent, child, depth, out, N);
}